// D3DispersionLayer_21191368639078
// MI455X (gfx1250) — compile-verified
//
#include <hip/hip_runtime.h>
#include <math.h>

// ---------------------------------------------------------------------------
// DFT-D3(BJ) dispersion energy, edge-parallel, for MI455X (gfx1250).
//
// Phase 0: zero nc[] (workspace) and out[].
// Phase 1: coordination numbers nc[i] = sum_e damp(e)  (atomic segment-sum).
// Phase 2: per-edge 5x5 reference-CN interpolation + BJ-damped -C6/r^6-C8/r^8,
//          atomic segment-sum into out[idx_i].
//
// MI455X reasoning (compile-only loop):
//  * HBM traffic ~25 MB -> ~1 us @ 23.3 TB/s; c6ab (2.7 MB) is L2-resident
//    (192 MB L2). The dominant memory op is the per-edge 300 B row gather
//    from c6ab viewed as a [9025 x 75] f32 table -- exactly the TDM
//    gather-mode pattern (CDNA5 ISA): each wave issues two
//    gather-mode tensor_load_to_lds ops (16 rows each, 16-bit indices built
//    from v_readlane of the lanes' pair IDs), waits on TENSORcnt, then reads
//    its rows from LDS (lane stride 75 dwords, gcd(75,64)=1 -> conflict-free).
//    This replaces ~19 VMEM loads/lane with one DMA per 16 lanes and frees
//    the VGPR load-landing space.
//  * rcov / r2r4 (95 f32) are hit by 2M random per-edge gathers; staged once
//    per workgroup into LDS via global_load_async_to_lds_b32 (ASYNCcnt).
//  * 25 exp/edge -> __expf -> v_exp_f32; branchless valid-masking (selects)
//    matches the reference (exp computed for all 25, masked in the sum).
//  * Segment sums via global_atomic_add_f32 (2M atomics over 50K addrs).
//  * No WMMA: no shared-operand matrix structure exists in this workload —
//    both interpolation operands are per-edge data.
// ---------------------------------------------------------------------------

#define D3_INV_AUTOANG (1.0f / 0.52917726f)
#define D3_K1 16.0f
#define D3_K3 -4.0f
#define D3_S6 1.0f
#define D3_S8 0.7875f
#define D3_A1 0.4289f
#define D3_A2 4.4407f
#define N_ELEM 95
#define MAXC 5
#define ROWLEN 75                     /* 25 refs * 3 floats               */
#define ROWBYTES (ROWLEN * 4)         /* 300 B                            */
#define NPAIR (N_ELEM * N_ELEM)       /* 9025 rows                        */
#define EDISP_BLK 128                 /* 4 wave32 -> 38.4 KB row slab     */
#define EDISP_WAVES (EDISP_BLK / 32)

typedef __attribute__((ext_vector_type(4))) unsigned int u32x4;
typedef __attribute__((ext_vector_type(8))) int          i32x8;
typedef __attribute__((ext_vector_type(4))) int          i32x4;

#define HAVE_TDM __has_builtin(__builtin_amdgcn_tensor_load_to_lds)

// --- gfx1250 async-copy helpers -------------------------------------------
__device__ __forceinline__ void async_load_dword_to_lds(const float* gaddr,
                                                        float* lds_ptr) {
    unsigned           lds = (unsigned)(uintptr_t)lds_ptr;  // low 32b = LDS byte addr
    unsigned long long ga  = (unsigned long long)(uintptr_t)gaddr;
    asm volatile("global_load_async_to_lds_b32 %0, %1, off"
                 :: "v"(lds), "v"(ga)
                 : "memory");
}

__device__ __forceinline__ void wait_asynccnt0() {
#if __has_builtin(__builtin_amdgcn_s_wait_asynccnt)
    __builtin_amdgcn_s_wait_asynccnt(0);
#else
    asm volatile("s_wait_asynccnt 0x0" ::: "memory");
#endif
}

__device__ __forceinline__ void wait_tensorcnt0() {
#if __has_builtin(__builtin_amdgcn_s_wait_tensorcnt)
    __builtin_amdgcn_s_wait_tensorcnt(0);
#else
    asm volatile("s_wait_tensorcnt 0x0" ::: "memory");
#endif
    asm volatile("" ::: "memory");    // keep LDS reads below the wait
}

__global__ void d3_zero_kernel(float* __restrict__ nc,
                               float* __restrict__ out,
                               int n_atoms) {
    int i = blockIdx.x * blockDim.x + threadIdx.x;
    if (i < n_atoms) {
        nc[i]  = 0.0f;
        out[i] = 0.0f;
    }
}

__global__ void d3_ncoord_kernel(const int* __restrict__ Z,
                                 const float* __restrict__ dist,
                                 const int* __restrict__ idx_i,
                                 const int* __restrict__ idx_j,
                                 const float* __restrict__ rcov,
                                 float* __restrict__ nc,
                                 int n_edges) {
    __shared__ float s_rcov[N_ELEM];
    int t = threadIdx.x;
    if (t < N_ELEM) {
        async_load_dword_to_lds(rcov + t, &s_rcov[t]);
    }
    wait_asynccnt0();
    __syncthreads();

    int e = blockIdx.x * blockDim.x + t;
    if (e >= n_edges) return;
    int ii = idx_i[e];
    int jj = idx_j[e];
    float r   = dist[e] * D3_INV_AUTOANG;               // Angstrom -> Bohr
    float rco = s_rcov[Z[ii]] + s_rcov[Z[jj]];
    float x    = -D3_K1 * (rco / r - 1.0f);
    float damp = 1.0f / (1.0f + __expf(x));             // 1/(1+exp(-k1*(rco/r-1)))
    atomicAdd(&nc[ii], damp);
}

__global__ void __launch_bounds__(EDISP_BLK)
d3_edisp_kernel(const int* __restrict__ Z,
                const float* __restrict__ dist,
                const int* __restrict__ idx_i,
                const int* __restrict__ idx_j,
                const float* __restrict__ r2r4,
                const float* __restrict__ c6ab,
                const float* __restrict__ nc,
                float* __restrict__ out,
                int n_edges) {
    __shared__ float s_r2r4[N_ELEM];
#if HAVE_TDM
    __shared__ float s_rows[EDISP_WAVES * 32 * ROWLEN];  // 300 B/lane slab
#endif

    int t = threadIdx.x;
    if (t < N_ELEM) {
        async_load_dword_to_lds(r2r4 + t, &s_r2r4[t]);
    }

    int e  = blockIdx.x * blockDim.x + t;
    // TDM ignores EXEC: every lane must contribute an in-bounds row index,
    // so tail lanes clamp; only the final atomic is guarded.
    int ec = (e < n_edges) ? e : (n_edges - 1);

    int ii = idx_i[ec];
    int jj = idx_j[ec];
    int Zi = Z[ii];
    int Zj = Z[jj];
    int pair = Zi * N_ELEM + Zj;                         // row in [9025 x 75]

#if HAVE_TDM
    int wave = t >> 5;
    // ---- build two gather-mode D# descriptors (16 rows each) ----
    // Group0: count=1 (b1:0), gather_index_size=0 (16-bit, b30),
    //         gather_mode=1 (b31), lds_addr (b63:32), global_addr (b120:64),
    //         type=2 (b127:126).
    unsigned lds_base = (unsigned)(uintptr_t)&s_rows[wave * (32 * ROWLEN)];
    unsigned lds0 = (unsigned)__builtin_amdgcn_readfirstlane((int)lds_base);
    unsigned long long ga = (unsigned long long)(uintptr_t)c6ab;
    unsigned ga_lo = (unsigned)ga;
    unsigned ga_hi = ((unsigned)(ga >> 32)) & 0x01FFFFFFu;
    u32x4 g0a = { 1u | (1u << 31), lds0,                  ga_lo, ga_hi | (2u << 30) };
    u32x4 g0b = { 1u | (1u << 31), lds0 + 16u * ROWBYTES, ga_lo, ga_hi | (2u << 30) };
    // Group1: workgroup_mask=0, data_size=2 (4B, b17:16), tensor_dim0=75
    // (b79:48), tensor_dim1=9025 (b111:80), tile_dim0=75 (b127:112),
    // tile_dim1=16 indices (b143:128), tensor_dim0_stride=75 (b207:160).
    i32x8 g1v = { 0x20000,
                  ROWLEN << 16,                 /* dim0 lo16 @ b63:48     */
                  (int)(NPAIR << 16),           /* dim0 hi=0, dim1 lo16   */
                  ROWLEN << 16,                 /* dim1 hi=0, tile_dim0   */
                  16,                           /* tile_dim1 = #indices   */
                  ROWLEN,                       /* dim0_stride lo32       */
                  0, 0 };
    // Groups 2/3: packed u16 row indices, lanes 0..15 / 16..31.
#define RLU16(L) (((unsigned)__builtin_amdgcn_readlane(pair, (L))) & 0xFFFFu)
    i32x4 g2a = { (int)(RLU16(0)  | (RLU16(1)  << 16)),
                  (int)(RLU16(2)  | (RLU16(3)  << 16)),
                  (int)(RLU16(4)  | (RLU16(5)  << 16)),
                  (int)(RLU16(6)  | (RLU16(7)  << 16)) };
    i32x4 g3a = { (int)(RLU16(8)  | (RLU16(9)  << 16)),
                  (int)(RLU16(10) | (RLU16(11) << 16)),
                  (int)(RLU16(12) | (RLU16(13) << 16)),
                  (int)(RLU16(14) | (RLU16(15) << 16)) };
    i32x4 g2b = { (int)(RLU16(16) | (RLU16(17) << 16)),
                  (int)(RLU16(18) | (RLU16(19) << 16)),
                  (int)(RLU16(20) | (RLU16(21) << 16)),
                  (int)(RLU16(22) | (RLU16(23) << 16)) };
    i32x4 g3b = { (int)(RLU16(24) | (RLU16(25) << 16)),
                  (int)(RLU16(26) | (RLU16(27) << 16)),
                  (int)(RLU16(28) | (RLU16(29) << 16)),
                  (int)(RLU16(30) | (RLU16(31) << 16)) };
#undef RLU16
    // amdgpu-toolchain (clang-23) 6-arg form: (g0, g1, g2, g3, extra v8i, cpol).
    // The 5th operand's semantics are uncharacterized; zero-filled per the
    // verified probe call.
    i32x8 gx = { 0, 0, 0, 0, 0, 0, 0, 0 };
    __builtin_amdgcn_tensor_load_to_lds(g0a, g1v, g2a, g3a, gx, 0);
    __builtin_amdgcn_tensor_load_to_lds(g0b, g1v, g2b, g3b, gx, 0);
#else
    const float* gbase = c6ab + (size_t)pair * ROWLEN;
    __builtin_prefetch(gbase,      0, 1);
    __builtin_prefetch(gbase + 32, 0, 1);
    __builtin_prefetch(gbase + 64, 0, 1);
#endif

    // Overlap the DMA with the remaining scalar gathers.
    float nci = nc[ii];
    float ncj = nc[jj];
    float r   = dist[ec] * D3_INV_AUTOANG;

    wait_asynccnt0();        // s_r2r4 staged
    __syncthreads();
#if HAVE_TDM
    wait_tensorcnt0();       // this wave's rows are in LDS
    const float* base = &s_rows[(t >> 5) * (32 * ROWLEN) + (t & 31) * ROWLEN];
#else
    const float* base = gbase;
#endif

    // 5x5 Gaussian interpolation over reference coordination numbers.
    float rsum  = 0.0f;
    float csum  = 0.0f;
    float minrr = INFINITY;
    float c6mem = 0.0f;
#pragma unroll
    for (int k = 0; k < MAXC * MAXC; ++k) {
        float c0 = base[3 * k + 0];
        float c1 = base[3 * k + 1];
        float c2 = base[3 * k + 2];
        float d1 = c1 - nci;
        float d2 = c2 - ncj;
        float rr = d1 * d1 + d2 * d2;
        // argmin, first-occurrence semantics (strict <), over ALL k.
        bool better = rr < minrr;
        c6mem = better ? c0 : c6mem;
        minrr = better ? rr : minrr;
        float w     = __expf(D3_K3 * rr);
        bool  valid = c0 > 0.0f;
        rsum += valid ? w : 0.0f;
        csum += valid ? w * c0 : 0.0f;
    }
    float c6 = (rsum > 0.0f) ? (csum / rsum) : c6mem;

    float rf = s_r2r4[Zi] * s_r2r4[Zj];
    float c8 = 3.0f * c6 * rf;

    // BJ damping.
    float tmp  = D3_A1 * sqrtf(c8 / c6) + D3_A2;
    float r2   = r * r;
    float r6   = r2 * r2 * r2;
    float r8   = r6 * r2;
    float t2   = tmp * tmp;
    float t6   = t2 * t2 * t2;
    float t8   = t6 * t2;
    float e6   = -0.5f * D3_S6 * c6 / (r6 + t6);
    float e8   = -0.5f * D3_S8 * c8 / (r8 + t8);

    if (e < n_edges) {
        atomicAdd(&out[ii], e6 + e8);
    }
}

extern "C" void kernel_launch(void* const* d_in, const int* in_sizes, int n_in,
                              void* d_out, int out_size, void* d_ws, size_t ws_size,
                              hipStream_t stream) {
    // setup_inputs() order: Z, edge_dist, edge_index, rcov, r2r4, c6ab
    const int*   Z     = (const int*)  d_in[0];
    const float* dist  = (const float*)d_in[1];
    const int*   eidx  = (const int*)  d_in[2];   // [2, n_edges] row-major
    const float* rcov  = (const float*)d_in[3];
    const float* r2r4  = (const float*)d_in[4];
    const float* c6ab  = (const float*)d_in[5];   // [95,95,5,5,3]

    int n_atoms = in_sizes[0];
    int n_edges = in_sizes[1];
    const int* idx_i = eidx;
    const int* idx_j = eidx + n_edges;

    float* out = (float*)d_out;
    float* nc  = (float*)d_ws;                    // n_atoms floats of scratch

    dim3 block256(256);
    d3_zero_kernel<<<(n_atoms + 255) / 256, block256, 0, stream>>>(nc, out, n_atoms);
    d3_ncoord_kernel<<<(n_edges + 255) / 256, block256, 0, stream>>>(
        Z, dist, idx_i, idx_j, rcov, nc, n_edges);

    dim3 blockE(EDISP_BLK);
    d3_edisp_kernel<<<(n_edges + EDISP_BLK - 1) / EDISP_BLK, blockE, 0, stream>>>(
        Z, dist, idx_i, idx_j, r2r4, c6ab, nc, out, n_edges);
}